// MRConv2d_343597384472
// MI455X (gfx1250) — compile-verified
//
#include <hip/hip_runtime.h>
#include <math.h>

// Problem constants (match reference)
constexpr int B_    = 8;
constexpr int C_    = 192;
constexpr int N_    = 4096;
constexpr int K_    = 16;
constexpr int G_    = 4;
constexpr int COUT_ = 384;
constexpr float EPS_ = 1e-5f;

typedef __attribute__((ext_vector_type(2))) float v2f;
typedef __attribute__((ext_vector_type(8))) float v8f;

// ---------------------------------------------------------------------------
// Kernel 1: transpose x (B,C,N) -> xT (B,N,C) so gathers read contiguous rows
// ---------------------------------------------------------------------------
__global__ __launch_bounds__(256) void transpose_k(const float* __restrict__ x,
                                                   float* __restrict__ xT) {
  __shared__ float tile[32][33];
  const int b  = blockIdx.z;
  const int n0 = blockIdx.x * 32;
  const int c0 = blockIdx.y * 32;
  const float* xb  = x  + (size_t)b * C_ * N_;
  float*       xtb = xT + (size_t)b * N_ * C_;
  #pragma unroll
  for (int i = 0; i < 4; ++i) {
    int c = c0 + threadIdx.y + 8 * i;
    tile[threadIdx.y + 8 * i][threadIdx.x] = xb[(size_t)c * N_ + n0 + threadIdx.x];
  }
  __syncthreads();
  #pragma unroll
  for (int i = 0; i < 4; ++i) {
    int n = n0 + threadIdx.y + 8 * i;
    xtb[(size_t)n * C_ + c0 + threadIdx.x] = tile[threadIdx.x][threadIdx.y + 8 * i];
  }
}

// ---------------------------------------------------------------------------
// Kernel 2: max-relative gather. Block = 192 threads (one per channel),
// handles (b, 32 consecutive n). Neighbor rows are contiguous 768B reads.
// Result staged in padded LDS and written back channel-major (coalesced).
// ---------------------------------------------------------------------------
__global__ __launch_bounds__(192) void gather_max_k(const float* __restrict__ xT,
                                                    const int* __restrict__ ei,
                                                    float* __restrict__ m) {
  __shared__ int   ji[32 * K_];
  __shared__ int   ii[32 * K_];
  __shared__ float mt[32][C_ + 1];   // +1 pad: bank-conflict-free transpose
  const int b   = blockIdx.y;
  const int n0  = blockIdx.x * 32;
  const int tid = threadIdx.x;       // 0..191 == channel

  const int* ej  = ei + ((size_t)b * N_ + n0) * K_;                      // edge_index[0] (j)
  const int* eii = ei + (size_t)B_ * N_ * K_ + ((size_t)b * N_ + n0) * K_; // edge_index[1] (i)
  for (int t = tid; t < 32 * K_; t += 192) { ji[t] = ej[t]; ii[t] = eii[t]; }
  __syncthreads();

  const float* xtb = xT + (size_t)b * N_ * C_;
  for (int nl = 0; nl < 32; ++nl) {
    float acc = -INFINITY;
    #pragma unroll
    for (int k = 0; k < K_; ++k) {
      int j = ji[nl * K_ + k];
      int i = ii[nl * K_ + k];
      acc = fmaxf(acc, xtb[(size_t)j * C_ + tid] - xtb[(size_t)i * C_ + tid]);
    }
    if (nl + 1 < 32)  // pull next n's first neighbor row toward the caches
      __builtin_prefetch(xtb + (size_t)ji[(nl + 1) * K_] * C_ + tid, 0, 0);
    mt[nl][tid] = acc;
  }
  __syncthreads();

  float* mb  = m + (size_t)b * C_ * N_;
  const int lane = tid & 31;
  const int grp  = tid >> 5;               // 6 groups of 32 lanes
  for (int c = grp; c < C_; c += 6)
    mb[(size_t)c * N_ + n0 + lane] = mt[lane][c];
}

// ---------------------------------------------------------------------------
// Kernel 3: grouped GEMM via V_WMMA_F32_16X16X4_F32.
// Grid: (N/128, 6 co-tiles, B*G). Block = 256 threads = 8 waves; wave w owns
// the 16x16 output tile at n0 = bx*128 + w*16. W-tile cached in LDS (stride 97).
// Groups 0,1 take the B-operand from x; groups 2,3 from m (the concat is free).
// ---------------------------------------------------------------------------
__global__ __launch_bounds__(256) void gemm_k(const float* __restrict__ x,
                                              const float* __restrict__ m,
                                              const float* __restrict__ W,
                                              const float* __restrict__ bias,
                                              float* __restrict__ y) {
  __shared__ float Wt[16 * 97];
  const int bg  = blockIdx.z;
  const int b   = bg >> 2;
  const int g   = bg & 3;
  const int coT = blockIdx.y;              // 0..5
  const int tid = threadIdx.x;

  const float* Wg = W + (size_t)(g * 96 + coT * 16) * 96;
  for (int e = tid; e < 16 * 96; e += 256) {
    int r = e / 96, c = e % 96;
    Wt[r * 97 + c] = Wg[(size_t)r * 96 + c];
  }
  __syncthreads();

  const int lane = tid & 31;
  const int wv   = tid >> 5;
  const int hi   = lane >> 4;              // selects K pair {0,1} vs {2,3}
  const int lo   = lane & 15;              // A: row; B: column; D: column
  const int n0   = blockIdx.x * 128 + wv * 16;

  const float* hs = (g < 2 ? x : m) + (size_t)b * C_ * N_ + (size_t)(g & 1) * 96 * N_;
  const float* hcol = hs + n0 + lo;

  v8f acc = {};
  #pragma unroll 4
  for (int kk = 0; kk < 96; kk += 4) {
    const int krow = kk + 2 * hi;
    v2f a;  a.x  = Wt[lo * 97 + krow];
            a.y  = Wt[lo * 97 + krow + 1];
    v2f bb; bb.x = hcol[(size_t)krow * N_];
            bb.y = hcol[(size_t)(krow + 1) * N_];
    acc = __builtin_amdgcn_wmma_f32_16x16x4_f32(false, a, false, bb,
                                                (short)0, acc, false, false);
  }

  float* yb = y + ((size_t)b * COUT_ + g * 96 + coT * 16) * N_;
  #pragma unroll
  for (int r = 0; r < 8; ++r) {
    const int row = r + 8 * hi;            // C/D layout: M = r + 8*(lane/16)
    const int cog = g * 96 + coT * 16 + row;
    yb[(size_t)row * N_ + n0 + lo] = acc[r] + bias[cog];
  }
}

// ---------------------------------------------------------------------------
// Kernel 4: per-channel sum / sumsq over (B,N). One block per (channel, b).
// ---------------------------------------------------------------------------
__global__ __launch_bounds__(256) void stats_k(const float* __restrict__ y,
                                               float* __restrict__ stats) {
  __shared__ float s1[256];
  __shared__ float s2[256];
  const int ch = blockIdx.x;
  const int b  = blockIdx.y;
  const float* row = y + ((size_t)b * COUT_ + ch) * N_;
  float sum = 0.f, sq = 0.f;
  for (int i = threadIdx.x; i < N_; i += 256) {
    float v = row[i];
    sum += v; sq += v * v;
  }
  s1[threadIdx.x] = sum; s2[threadIdx.x] = sq;
  __syncthreads();
  for (int off = 128; off > 0; off >>= 1) {
    if (threadIdx.x < off) {
      s1[threadIdx.x] += s1[threadIdx.x + off];
      s2[threadIdx.x] += s2[threadIdx.x + off];
    }
    __syncthreads();
  }
  if (threadIdx.x == 0) {
    atomicAdd(&stats[ch], s1[0]);
    atomicAdd(&stats[COUT_ + ch], s2[0]);
  }
}

// ---------------------------------------------------------------------------
// Kernel 5: fold mean/var/gamma/beta into per-channel scale+shift.
// ---------------------------------------------------------------------------
__global__ void finalize_k(const float* __restrict__ stats,
                           const float* __restrict__ gamma,
                           const float* __restrict__ beta,
                           float* __restrict__ ss) {
  const int ch = threadIdx.x;
  const float inv  = 1.0f / (float)(B_ * N_);
  const float mean = stats[ch] * inv;
  const float var  = stats[COUT_ + ch] * inv - mean * mean;  // biased variance
  const float sc   = gamma[ch] * rsqrtf(var + EPS_);
  ss[ch]          = sc;
  ss[COUT_ + ch]  = beta[ch] - mean * sc;
}

// ---------------------------------------------------------------------------
// Kernel 6: normalize + exact GELU (erf form, matches approximate=False).
// ---------------------------------------------------------------------------
__global__ __launch_bounds__(256) void act_k(const float* __restrict__ y,
                                             const float* __restrict__ ss,
                                             float* __restrict__ out) {
  size_t idx = ((size_t)blockIdx.x * 256 + threadIdx.x) * 8;
  const int ch = (int)((idx / N_) % COUT_);   // 8 consecutive stay in one row
  const float sc = ss[ch], sh = ss[COUT_ + ch];
  #pragma unroll
  for (int i = 0; i < 8; ++i) {
    float v = y[idx + i] * sc + sh;
    out[idx + i] = 0.5f * v * (1.0f + erff(v * 0.70710678118654752f));
  }
}

// ---------------------------------------------------------------------------
extern "C" void kernel_launch(void* const* d_in, const int* in_sizes, int n_in,
                              void* d_out, int out_size, void* d_ws, size_t ws_size,
                              hipStream_t stream) {
  const float* x     = (const float*)d_in[0];
  const int*   ei    = (const int*)d_in[1];
  const float* W     = (const float*)d_in[2];
  const float* bias  = (const float*)d_in[3];
  const float* gamma = (const float*)d_in[4];
  const float* beta  = (const float*)d_in[5];
  float* out = (float*)d_out;

  float* ws    = (float*)d_ws;
  float* xT    = ws;                                   // B*N*C      = 6,291,456
  float* m     = ws + (size_t)B_ * N_ * C_;            // B*C*N      = 6,291,456
  float* y     = m  + (size_t)B_ * C_ * N_;            // B*COUT*N   = 12,582,912
  float* stats = y  + (size_t)B_ * COUT_ * N_;         // 2*COUT
  float* ss    = stats + 2 * COUT_;                    // 2*COUT

  hipMemsetAsync(stats, 0, 2 * COUT_ * sizeof(float), stream);

  transpose_k<<<dim3(N_ / 32, C_ / 32, B_), dim3(32, 8), 0, stream>>>(x, xT);
  gather_max_k<<<dim3(N_ / 32, B_), 192, 0, stream>>>(xT, ei, m);
  gemm_k<<<dim3(N_ / 128, 6, B_ * G_), 256, 0, stream>>>(x, m, W, bias, y);
  stats_k<<<dim3(COUT_, B_), 256, 0, stream>>>(y, stats);
  finalize_k<<<1, COUT_, 0, stream>>>(stats, gamma, beta, ss);
  act_k<<<(B_ * COUT_ * N_) / (256 * 8), 256, 0, stream>>>(y, ss, out);
}